// Group_15333033247025
// MI455X (gfx1250) — compile-verified
//
#include <hip/hip_runtime.h>

// Point-MAE Group for MI455X (gfx1250, wave32):
//   FPS: 1 WG/batch, points staged into LDS (SoA) with async global->LDS
//        copies (ASYNCcnt), dist in regs, wave shfl_xor argmax chain.
//   KNN: distance tiles via V_WMMA_F32_16X16X4_F32 (K=4 homogeneous trick:
//        A=(-2c,1), B=(x,|x|^2) -> D = -2 c.x + |x|^2 ; per-row |c|^2 constant
//        dropped since it cannot change per-row top-k ordering).
//   Selection: per-wave sorted top-32 lists in LDS + ballot-gated insertion,
//        then 8-way merge with wave argmin rounds; LDS gather for output.

#define NPTS   8192
#define NGRP   512
#define GSZ    32
#define BATCH  32
#define TPB    256
#define PAD    8208   // 8192 + 16 padding to break LDS bank alignment

typedef float v2f __attribute__((ext_vector_type(2)));
typedef float v8f __attribute__((ext_vector_type(8)));

typedef __attribute__((address_space(1))) int* gas_i32p;  // global (AS1)
typedef __attribute__((address_space(3))) int* las_i32p;  // LDS (AS3)

#if defined(__has_builtin)
#if __has_builtin(__builtin_amdgcn_global_load_async_to_lds_b32)
#define HAVE_ASYNC_LDS 1
#endif
#if __has_builtin(__builtin_amdgcn_s_wait_asynccnt)
#define HAVE_WAIT_ASYNC 1
#endif
#endif

static __device__ __forceinline__ float f_inf() { return __builtin_inff(); }

static __device__ __forceinline__ void wait_async0() {
#if defined(HAVE_ASYNC_LDS)
#if defined(HAVE_WAIT_ASYNC)
  __builtin_amdgcn_s_wait_asynccnt(0);
#else
  asm volatile("s_wait_asynccnt 0" ::: "memory");
#endif
#endif
}

// Stage one batch's points global->LDS as SoA (x,y,z arrays).
// Async path: per-lane b32 async-to-LDS copies; global side stays coalesced
// (consecutive lanes read consecutive dwords), LDS side scatters the
// 3-strided transpose. Bypasses VGPRs entirely on gfx1250.
static __device__ __forceinline__ void stage_points(const float* __restrict__ src,
                                                    float* sx, float* sy, float* sz,
                                                    int t) {
#if defined(HAVE_ASYNC_LDS)
  for (int i = t; i < NPTS * 3; i += TPB) {
    const int p = i / 3;
    const int comp = i - 3 * p;
    float* dst = (comp == 0) ? (sx + p) : (comp == 1) ? (sy + p) : (sz + p);
    __builtin_amdgcn_global_load_async_to_lds_b32(
        (gas_i32p)(src + i), (las_i32p)dst, 0, 0);
  }
  wait_async0();
#else
  for (int p = t; p < NPTS; p += TPB) {
    sx[p] = src[p * 3 + 0];
    sy[p] = src[p * 3 + 1];
    sz[p] = src[p * 3 + 2];
  }
#endif
}

// Insert (v, idx) into a sorted-ascending 32-entry (dist, idx) list.
// Equal distances keep earlier (lower) index first -> matches lax.top_k ties.
static __device__ void insert32(float* td, int* ti, float v, int idx) {
  if (!(v < td[31])) return;
  int pos = 32;
  for (int k = 0; k < 32; ++k) {
    if (v < td[k]) { pos = k; break; }
  }
  for (int k = 31; k > pos; --k) { td[k] = td[k - 1]; ti[k] = ti[k - 1]; }
  td[pos] = v; ti[pos] = idx;
}

// ---------------------------------------------------------------------------
// Kernel 1: furthest point sampling. One workgroup per batch element.
// ---------------------------------------------------------------------------
__global__ __launch_bounds__(TPB) void fps_kernel(const float* __restrict__ xyz,
                                                  float* __restrict__ centers /*[B,G,3]*/) {
  extern __shared__ float smem[];
  float* sx   = smem;                 // [PAD]
  float* sy   = sx + PAD;             // [PAD]
  float* sz   = sy + PAD;             // [PAD]
  float* bc   = sz + PAD;             // [4]  broadcast centroid
  float* redD = bc + 4;               // [8]  per-wave argmax dist
  int*   redI = (int*)(redD + 8);     // [8]  per-wave argmax idx

  const int b = blockIdx.x;
  const int t = threadIdx.x;
  const int lane = t & 31, wv = t >> 5;
  const float* src = xyz + (size_t)b * NPTS * 3;

  stage_points(src, sx, sy, sz, t);
  __syncthreads();

  if (t == 0) {
    // deterministic start at index 0 (matches reference)
    bc[0] = sx[0]; bc[1] = sy[0]; bc[2] = sz[0];
    float* c0 = centers + ((size_t)b * NGRP) * 3;
    c0[0] = bc[0]; c0[1] = bc[1]; c0[2] = bc[2];
  }
  __syncthreads();

  float dist[32];
#pragma unroll
  for (int j = 0; j < 32; ++j) dist[j] = f_inf();

  for (int it = 1; it < NGRP; ++it) {
    const float cx = bc[0], cy = bc[1], cz = bc[2];
    float bd = -1.0f; int bi = 0x7FFFFFFF;
#pragma unroll
    for (int j = 0; j < 32; ++j) {
      const int p = j * TPB + t;            // interleaved ownership
      const float dx = sx[p] - cx, dy = sy[p] - cy, dz = sz[p] - cz;
      const float d  = dx * dx + dy * dy + dz * dz;
      const float nd = fminf(dist[j], d);
      dist[j] = nd;
      if (nd > bd) { bd = nd; bi = p; }     // strict > : first max wins
    }
    // wave32 argmax reduce, lower index wins ties
#pragma unroll
    for (int off = 16; off >= 1; off >>= 1) {
      const float od = __shfl_xor(bd, off, 32);
      const int   oi = __shfl_xor(bi, off, 32);
      if (od > bd || (od == bd && oi < bi)) { bd = od; bi = oi; }
    }
    if (lane == 0) { redD[wv] = bd; redI[wv] = bi; }
    __syncthreads();
    if (t < 8) {
      float bd2 = redD[t]; int bi2 = redI[t];
#pragma unroll
      for (int off = 4; off >= 1; off >>= 1) {
        const float od = __shfl_xor(bd2, off, 32);
        const int   oi = __shfl_xor(bi2, off, 32);
        if (od > bd2 || (od == bd2 && oi < bi2)) { bd2 = od; bi2 = oi; }
      }
      if (t == 0) {
        const float nx = sx[bi2], ny = sy[bi2], nz = sz[bi2];
        bc[0] = nx; bc[1] = ny; bc[2] = nz;
        float* co = centers + ((size_t)b * NGRP + it) * 3;
        co[0] = nx; co[1] = ny; co[2] = nz;
      }
    }
    __syncthreads();
  }
}

// ---------------------------------------------------------------------------
// Kernel 2: KNN (WMMA distance tiles) + top-32 select + gather.
// One workgroup (8 waves) per (batch, 16-center tile). Grid 32x32.
// ---------------------------------------------------------------------------
__global__ __launch_bounds__(TPB) void knn_kernel(const float* __restrict__ xyz,
                                                  const float* __restrict__ centers,
                                                  float* __restrict__ nbr /*[B,G,K,3]*/) {
  extern __shared__ float smem[];
  float* sx = smem;                   // [PAD]
  float* sy = sx + PAD;               // [PAD]
  float* sz = sy + PAD;               // [PAD]
  float* sr = sz + PAD;               // [PAD]  |x|^2
  float* Wd = sr + PAD;               // [8][16][32] per-wave top dists
  int*   Wi = (int*)(Wd + 8 * 16 * 32);   // [8][16][32] per-wave top idx
  int*   Fi = Wi + 8 * 16 * 32;       // [16][32] final idx
  float* Cx = (float*)(Fi + 16 * 32); // [16]
  float* Cy = Cx + 16;                // [16]
  float* Cz = Cy + 16;                // [16]

  const int b    = blockIdx.y;
  const int g0   = blockIdx.x * 16;
  const int t    = threadIdx.x;
  const int lane = t & 31, wv = t >> 5;
  const bool hi  = lane >= 16;
  const int  m   = lane & 15;

  const float* src = xyz + (size_t)b * NPTS * 3;
  stage_points(src, sx, sy, sz, t);
  if (t < 16) {
    const float* c = centers + ((size_t)b * NGRP + g0 + t) * 3;
    Cx[t] = c[0]; Cy[t] = c[1]; Cz[t] = c[2];
  }
  for (int e = t; e < 8 * 16 * 32; e += TPB) { Wd[e] = f_inf(); Wi[e] = 0x7FFFFFFF; }
  __syncthreads();

  // |x|^2 from staged SoA
  for (int p = t; p < NPTS; p += TPB) {
    const float x = sx[p], y = sy[p], z = sz[p];
    sr[p] = x * x + y * y + z * z;
  }
  __syncthreads();

  // A-matrix (16x4 f32, 2 VGPRs; loop-invariant):
  //  lanes 0-15 hold (K=0,K=1) = (-2cx, -2cy); lanes 16-31 hold (K=2,K=3) = (-2cz, 1)
  v2f A;
  A.x = hi ? (-2.0f * Cz[m]) : (-2.0f * Cx[m]);
  A.y = hi ? 1.0f            : (-2.0f * Cy[m]);

  float* myWd = Wd + (wv * 16) * 32;
  int*   myWi = Wi + (wv * 16) * 32;

  for (int iter = 0; iter < NPTS / 128; ++iter) {       // 64 tiles per wave
    const int nb = iter * 128 + wv * 16;                // disjoint columns per wave
    const int c  = nb + m;
    // B-matrix (4x16 f32, 2 VGPRs): rows (x, y) in lanes 0-15, (z, |x|^2) in 16-31
    v2f Bv;
    Bv.x = hi ? sz[c] : sx[c];
    Bv.y = hi ? sr[c] : sy[c];
    v8f acc = {0.f, 0.f, 0.f, 0.f, 0.f, 0.f, 0.f, 0.f};
    acc = __builtin_amdgcn_wmma_f32_16x16x4_f32(false, A, false, Bv,
                                                (short)0, acc, false, false);
    // D layout: row = r + 8*(lane>=16), col = nb + (lane&15)
#pragma unroll
    for (int r = 0; r < 8; ++r) {
      const float v = acc[r];
      const int myRow = r + (hi ? 8 : 0);
      const float th = myWd[myRow * 32 + 31];
      const unsigned pass = __builtin_amdgcn_ballot_w32(v < th);
      if (pass) {                                        // uniform branch
        for (int s = 0; s < 16; ++s) {
          const float vLo = __shfl(v, s,      32);
          const float vHi = __shfl(v, s + 16, 32);
          if (lane == 0  && ((pass >> s) & 1u))
            insert32(myWd + r * 32,       myWi + r * 32,       vLo, nb + s);
          if (lane == 16 && ((pass >> (s + 16)) & 1u))
            insert32(myWd + (r + 8) * 32, myWi + (r + 8) * 32, vHi, nb + s);
        }
      }
    }
  }
  __syncthreads();

  // Merge 8 per-wave sorted lists per row -> global smallest 32 (wave per 2 rows)
  for (int rr = 0; rr < 2; ++rr) {
    const int row = wv * 2 + rr;
    float cd[8]; int ci[8];
#pragma unroll
    for (int q = 0; q < 8; ++q) {
      const int e = lane * 8 + q;                        // 0..255 candidates
      cd[q] = Wd[((e >> 5) * 16 + row) * 32 + (e & 31)];
      ci[q] = Wi[((e >> 5) * 16 + row) * 32 + (e & 31)];
    }
    for (int sel = 0; sel < 32; ++sel) {
      float bd = f_inf(); int bi = 0x7FFFFFFF; int bq = -1;
#pragma unroll
      for (int q = 0; q < 8; ++q)
        if (cd[q] < bd || (cd[q] == bd && ci[q] < bi)) { bd = cd[q]; bi = ci[q]; bq = q; }
      float gbd = bd; int gbi = bi;
#pragma unroll
      for (int off = 16; off >= 1; off >>= 1) {
        const float od = __shfl_xor(gbd, off, 32);
        const int   oi = __shfl_xor(gbi, off, 32);
        if (od < gbd || (od == gbd && oi < gbi)) { gbd = od; gbi = oi; }
      }
      if (bq >= 0 && bd == gbd && bi == gbi) cd[bq] = f_inf();  // retire winner
      if (lane == 0) Fi[row * 32 + sel] = gbi;
    }
  }
  __syncthreads();

  // Gather neighborhoods, subtract center, write out
  for (int e = t; e < 16 * GSZ; e += TPB) {
    const int row = e >> 5, k = e & 31;
    const int idx = Fi[row * 32 + k];
    const int g = g0 + row;
    float* o = nbr + (((size_t)b * NGRP + g) * GSZ + k) * 3;
    o[0] = sx[idx] - Cx[row];
    o[1] = sy[idx] - Cy[row];
    o[2] = sz[idx] - Cz[row];
  }
}

// ---------------------------------------------------------------------------
extern "C" void kernel_launch(void* const* d_in, const int* in_sizes, int n_in,
                              void* d_out, int out_size, void* d_ws, size_t ws_size,
                              hipStream_t stream) {
  (void)in_sizes; (void)n_in; (void)d_ws; (void)ws_size; (void)out_size;
  const float* xyz = (const float*)d_in[0];
  float* out     = (float*)d_out;
  float* nbr     = out;                                         // [32,512,32,3]
  float* centers = out + (size_t)BATCH * NGRP * GSZ * 3;        // [32,512,3]

  const size_t fpsLds = (size_t)(3 * PAD + 4 + 8) * sizeof(float) + 8 * sizeof(int);
  fps_kernel<<<BATCH, TPB, fpsLds, stream>>>(xyz, centers);

  const size_t knnLds = (size_t)(4 * PAD) * sizeof(float)
                      + (size_t)(8 * 16 * 32) * (sizeof(float) + sizeof(int))
                      + (size_t)(16 * 32) * sizeof(int)
                      + (size_t)(3 * 16) * sizeof(float);
  knn_kernel<<<dim3(NGRP / 16, BATCH), TPB, knnLds, stream>>>(xyz, centers, nbr);
}